// DynamicBallQuery_18262200942681
// MI455X (gfx1250) — compile-verified
//
#include <hip/hip_runtime.h>

// DynamicBallQuery on MI455X (gfx1250, wave32).
// Distances via V_WMMA_F32_16X16X4_F32 using ||p-c||^2 = (|p|^2 - 2 p.c) + |c|^2.
// Each wave owns 32 centers; streams 32 points/iter => 4 WMMAs per chunk.

typedef __attribute__((ext_vector_type(2))) float v2f;
typedef __attribute__((ext_vector_type(8))) float v8f;

#define MIN_RADIUS 0.05f
#define MAX_RADIUS 0.3f
#define KNN 16
#define B_ 4
#define N_ 16384
#define M_ 2048
#define C_ 64

__device__ __forceinline__ float lanef(float v, int src) { return __shfl(v, src, 32); }

__device__ __forceinline__ float inv_vol_const() {
    return 1.0f / ((4.0f / 3.0f) * 3.14159265358979323846f *
                   MIN_RADIUS * MIN_RADIUS * MIN_RADIUS + 1e-8f);
}

// ---------------------------------------------------------------------------
// Pass 1: per-center count of points with dist < MIN_RADIUS (WMMA distances)
// ---------------------------------------------------------------------------
__global__ void __launch_bounds__(128)
count_kernel(const float* __restrict__ points,
             const int* __restrict__ cidx,
             float* __restrict__ counts) {
    const int lane = threadIdx.x & 31;
    const int wave = threadIdx.x >> 5;
    const int w = blockIdx.x * 4 + wave;       // 0..255 waves total
    const int cbase = w * 32;                  // flat center base (b-major)
    const int b = cbase / M_;
    const int m = (cbase % M_) + lane;
    const bool lo = lane < 16;

    const float* P = points + (size_t)b * N_ * 3;
    const int ci = cidx[b * M_ + m];
    const float cx = P[ci * 3 + 0], cy = P[ci * 3 + 1], cz = P[ci * 3 + 2];

    // Build two B tiles (centers 0..15 and 16..31 of this wave).
    const int s0 = (lane & 15);
    const int s1 = 16 + (lane & 15);
    const float c0x = lanef(cx, s0), c0y = lanef(cy, s0), c0z = lanef(cz, s0);
    const float c1x = lanef(cx, s1), c1y = lanef(cy, s1), c1z = lanef(cz, s1);
    v2f b0, b1;
    b0.x = lo ? (-2.0f * c0x) : (-2.0f * c0z);
    b0.y = lo ? (-2.0f * c0y) : 1.0f;
    b1.x = lo ? (-2.0f * c1x) : (-2.0f * c1z);
    b1.y = lo ? (-2.0f * c1y) : 1.0f;
    const float cc0 = c0x * c0x + c0y * c0y + c0z * c0z;
    const float cc1 = c1x * c1x + c1y * c1y + c1z * c1z;

    const float r2min = MIN_RADIUS * MIN_RADIUS;
    float cnt0 = 0.0f, cnt1 = 0.0f;

    for (int nb = 0; nb < N_; nb += 32) {
        const int q0 = nb + (lane & 15);
        const int q1 = q0 + 16;
        if (nb + 64 < N_) __builtin_prefetch(P + (size_t)(q0 + 64) * 3, 0, 1);
        const float p0x = P[q0 * 3 + 0], p0y = P[q0 * 3 + 1], p0z = P[q0 * 3 + 2];
        const float p1x = P[q1 * 3 + 0], p1y = P[q1 * 3 + 1], p1z = P[q1 * 3 + 2];
        const float pp0 = p0x * p0x + p0y * p0y + p0z * p0z;
        const float pp1 = p1x * p1x + p1y * p1y + p1z * p1z;
        v2f a0, a1;
        a0.x = lo ? p0x : p0z;  a0.y = lo ? p0y : pp0;
        a1.x = lo ? p1x : p1z;  a1.y = lo ? p1y : pp1;
        v8f z = {};
        v8f c00 = __builtin_amdgcn_wmma_f32_16x16x4_f32(false, a0, false, b0, (short)0, z, false, false);
        v8f c01 = __builtin_amdgcn_wmma_f32_16x16x4_f32(false, a0, false, b1, (short)0, z, false, false);
        v8f c10 = __builtin_amdgcn_wmma_f32_16x16x4_f32(false, a1, false, b0, (short)0, z, false, false);
        v8f c11 = __builtin_amdgcn_wmma_f32_16x16x4_f32(false, a1, false, b1, (short)0, z, false, false);
#pragma unroll
        for (int r = 0; r < 8; ++r) {
            cnt0 += (c00[r] + cc0 < r2min) ? 1.0f : 0.0f;
            cnt1 += (c01[r] + cc1 < r2min) ? 1.0f : 0.0f;
            cnt0 += (c10[r] + cc0 < r2min) ? 1.0f : 0.0f;
            cnt1 += (c11[r] + cc1 < r2min) ? 1.0f : 0.0f;
        }
    }
    cnt0 += __shfl_xor(cnt0, 16, 32);
    cnt1 += __shfl_xor(cnt1, 16, 32);
    counts[cbase + lane] = lo ? cnt0 : cnt1;
}

// ---------------------------------------------------------------------------
// Pass 2: global density max (scalar)
// ---------------------------------------------------------------------------
__global__ void __launch_bounds__(256)
maxred_kernel(const float* __restrict__ counts, float* __restrict__ dmax) {
    __shared__ float red[256];
    float mx = 0.0f;
    for (int i = threadIdx.x; i < B_ * M_; i += 256) mx = fmaxf(mx, counts[i]);
    red[threadIdx.x] = mx;
    __syncthreads();
    for (int s = 128; s > 0; s >>= 1) {
        if (threadIdx.x < s) red[threadIdx.x] = fmaxf(red[threadIdx.x], red[threadIdx.x + s]);
        __syncthreads();
    }
    if (threadIdx.x == 0) dmax[0] = red[0] * inv_vol_const() + 1e-8f;
}

// ---------------------------------------------------------------------------
// Pass 3: WMMA distances + register-resident top-16 per center
// ---------------------------------------------------------------------------
__global__ void __launch_bounds__(128)
topk_kernel(const float* __restrict__ points,
            const int* __restrict__ cidx,
            const float* __restrict__ counts,
            const float* __restrict__ dmaxp,
            int* __restrict__ topk) {
    __shared__ float lds[4 * 32 * 33];          // 4 waves, 32 centers x 32 pts (pad 33)
    const int lane = threadIdx.x & 31;
    const int wave = threadIdx.x >> 5;
    float* wl = lds + wave * (32 * 33);
    const int w = blockIdx.x * 4 + wave;
    const int cbase = w * 32;
    const int b = cbase / M_;
    const int m = (cbase % M_) + lane;
    const bool lo = lane < 16;

    const float* P = points + (size_t)b * N_ * 3;
    const int ci = cidx[b * M_ + m];
    const float cx = P[ci * 3 + 0], cy = P[ci * 3 + 1], cz = P[ci * 3 + 2];

    // per-lane (per-center) radius^2
    const float dens = counts[cbase + lane] * inv_vol_const();
    const float rad = MIN_RADIUS + (MAX_RADIUS - MIN_RADIUS) * (1.0f - dens / dmaxp[0]);
    const float r2 = rad * rad;

    const int s0 = (lane & 15);
    const int s1 = 16 + (lane & 15);
    const float c0x = lanef(cx, s0), c0y = lanef(cy, s0), c0z = lanef(cz, s0);
    const float c1x = lanef(cx, s1), c1y = lanef(cy, s1), c1z = lanef(cz, s1);
    v2f b0, b1;
    b0.x = lo ? (-2.0f * c0x) : (-2.0f * c0z);
    b0.y = lo ? (-2.0f * c0y) : 1.0f;
    b1.x = lo ? (-2.0f * c1x) : (-2.0f * c1z);
    b1.y = lo ? (-2.0f * c1y) : 1.0f;
    const float cc0 = c0x * c0x + c0y * c0y + c0z * c0z;
    const float cc1 = c1x * c1x + c1y * c1y + c1z * c1z;

    float dl[16]; int il[16];
#pragma unroll
    for (int j = 0; j < 16; ++j) { dl[j] = 3.0e38f; il[j] = 0; }

    const int mycol = lane * 33;
    const int wcol0 = ((lane & 15) + 0) * 33 + (lo ? 0 : 8);
    const int wcol1 = ((lane & 15) + 16) * 33 + (lo ? 0 : 8);

    for (int nb = 0; nb < N_; nb += 32) {
        const int q0 = nb + (lane & 15);
        const int q1 = q0 + 16;
        if (nb + 64 < N_) __builtin_prefetch(P + (size_t)(q0 + 64) * 3, 0, 1);
        const float p0x = P[q0 * 3 + 0], p0y = P[q0 * 3 + 1], p0z = P[q0 * 3 + 2];
        const float p1x = P[q1 * 3 + 0], p1y = P[q1 * 3 + 1], p1z = P[q1 * 3 + 2];
        const float pp0 = p0x * p0x + p0y * p0y + p0z * p0z;
        const float pp1 = p1x * p1x + p1y * p1y + p1z * p1z;
        v2f a0, a1;
        a0.x = lo ? p0x : p0z;  a0.y = lo ? p0y : pp0;
        a1.x = lo ? p1x : p1z;  a1.y = lo ? p1y : pp1;
        v8f z = {};
        v8f c00 = __builtin_amdgcn_wmma_f32_16x16x4_f32(false, a0, false, b0, (short)0, z, false, false);
        v8f c01 = __builtin_amdgcn_wmma_f32_16x16x4_f32(false, a0, false, b1, (short)0, z, false, false);
        v8f c10 = __builtin_amdgcn_wmma_f32_16x16x4_f32(false, a1, false, b0, (short)0, z, false, false);
        v8f c11 = __builtin_amdgcn_wmma_f32_16x16x4_f32(false, a1, false, b1, (short)0, z, false, false);

        // transpose through LDS: row = center (0..31), col = point (0..31)
#pragma unroll
        for (int r = 0; r < 8; ++r) {
            wl[wcol0 + 0  + r] = c00[r] + cc0;
            wl[wcol1 + 0  + r] = c01[r] + cc1;
            wl[wcol0 + 16 + r] = c10[r] + cc0;
            wl[wcol1 + 16 + r] = c11[r] + cc1;
        }
        __syncthreads();

        for (int j = 0; j < 32; ++j) {
            const float v = wl[mycol + j];
            const float cd = (v < r2) ? v : 1e10f;   // squared-space mask, same ordering
            const int cq = nb + j;
            if (cd < dl[15]) {                        // sorted-insert, stable ties
#pragma unroll
                for (int u = 15; u >= 1; --u) {
                    const bool sh  = cd < dl[u - 1];
                    const bool ins = (cd < dl[u]) && !sh;
                    dl[u] = sh ? dl[u - 1] : (ins ? cd : dl[u]);
                    il[u] = sh ? il[u - 1] : (ins ? cq : il[u]);
                }
                if (cd < dl[0]) { dl[0] = cd; il[0] = cq; }
            }
        }
        __syncthreads();
    }

    int* outp = topk + (size_t)(cbase + lane) * KNN;
#pragma unroll
    for (int j = 0; j < 16; ++j) outp[j] = il[j];
}

// ---------------------------------------------------------------------------
// Pass 4: gather neighbor features, 64 floats per (center, neighbor) row
// ---------------------------------------------------------------------------
__global__ void __launch_bounds__(256)
gather_kernel(const float* __restrict__ features,
              const int* __restrict__ topk,
              float* __restrict__ out) {
    const size_t row = (size_t)blockIdx.x * 4 + (threadIdx.x >> 6);  // (b*M+m)*K + k
    const int c = threadIdx.x & 63;
    int idx = topk[row];
    idx = idx < 0 ? 0 : (idx > N_ - 1 ? N_ - 1 : idx);
    const int b = (int)(row / ((size_t)M_ * KNN));
    out[row * C_ + c] = features[((size_t)b * N_ + idx) * C_ + c];
}

// ---------------------------------------------------------------------------
extern "C" void kernel_launch(void* const* d_in, const int* in_sizes, int n_in,
                              void* d_out, int out_size, void* d_ws, size_t ws_size,
                              hipStream_t stream) {
    const float* points   = (const float*)d_in[0];   // [B,N,3]
    const float* features = (const float*)d_in[1];   // [B,N,C]
    const int*   cidx     = (const int*)d_in[2];     // [B,M]
    float* out = (float*)d_out;                      // [B,M,K,C]

    float* wsf    = (float*)d_ws;
    float* dmax   = wsf;                             // 1 float (padded to 16)
    float* counts = wsf + 16;                        // B*M floats
    int*   topk   = (int*)(wsf + 16 + B_ * M_);      // B*M*K ints

    hipLaunchKernelGGL(count_kernel, dim3((B_ * M_) / 128), dim3(128), 0, stream,
                       points, cidx, counts);
    hipLaunchKernelGGL(maxred_kernel, dim3(1), dim3(256), 0, stream, counts, dmax);
    hipLaunchKernelGGL(topk_kernel, dim3((B_ * M_) / 128), dim3(128), 0, stream,
                       points, cidx, counts, dmax, topk);
    hipLaunchKernelGGL(gather_kernel, dim3((B_ * M_ * KNN) / 4), dim3(256), 0, stream,
                       features, topk, out);
}